// SSIM_20478404067651
// MI455X (gfx1250) — compile-verified
//
#include <hip/hip_runtime.h>
#include <math.h>

// ---------------------------------------------------------------------------
// SSIM (11x11 Gaussian, sigma=1.5) on 16x3x512x512 fp32 via f32 WMMA band-GEMM
// + async global->LDS staging (CDNA5 ASYNCcnt path) for interior tiles
// ---------------------------------------------------------------------------

typedef float v2f __attribute__((ext_vector_type(2)));
typedef float v8f __attribute__((ext_vector_type(8)));
typedef int   v4i __attribute__((vector_size(16)));

#define IMG_W   512
#define IMG_H   512
#define TILE_Y  32          // block output tile rows
#define TILE_X  64          // block output tile cols
#define SROWS   44          // staged rows  (32 + 5 halo + K-band pad to 27)
#define SW      88          // staged cols  (origin x0-8, 16B-aligned, 64+halo+pad)
#define XOFF    8           // left halo (vs 5 in math; +3 column shift below)
#define SSIM_C1 1e-4f
#define SSIM_C2 9e-4f

#define GLOBAL_AS __attribute__((address_space(1)))
#define LDS_AS    __attribute__((address_space(3)))

__device__ __forceinline__ v8f wmma4(v2f a, v2f b, v8f c) {
  // D = A(16x4,f32) x B(4x16,f32) + C(16x16,f32)
  return __builtin_amdgcn_wmma_f32_16x16x4_f32(
      /*neg_a=*/false, a, /*neg_b=*/false, b,
      /*c_mod=*/(short)0, c, /*reuse_a=*/false, /*reuse_b=*/false);
}

__device__ __forceinline__ void async_copy16(const float* g, float* l) {
#if __has_builtin(__builtin_amdgcn_global_load_async_to_lds_b128)
  __builtin_amdgcn_global_load_async_to_lds_b128(
      (GLOBAL_AS v4i*)g, (LDS_AS v4i*)l, /*offset=*/0, /*cpol=*/0);
#else
  *(float4*)l = *(const float4*)g;
#endif
}

__device__ __forceinline__ void wait_async() {
#if __has_builtin(__builtin_amdgcn_s_wait_asynccnt)
  __builtin_amdgcn_s_wait_asynccnt(0);
#endif
}

__global__ __launch_bounds__(256) void ssim_conv_kernel(
    const float* __restrict__ img1, const float* __restrict__ img2,
    float* __restrict__ partial) {
  __shared__ alignas(16) float S1[SROWS * SW];  // staged img1 tile (+halo)
  __shared__ alignas(16) float S2[SROWS * SW];  // staged img2 tile
  __shared__ float scrT[8 * 16 * 32];           // per-wave C->A layout bounce
  __shared__ float gtab[11];                    // normalized gaussian taps
  __shared__ float red[256];                    // block reduction

  const int tid = threadIdx.x;

  // ---- block -> (plane, tile) mapping: 128 tiles per 512x512 plane ----
  const int p  = blockIdx.x >> 7;       // plane 0..47 (n*3 + c)
  const int t  = blockIdx.x & 127;
  const int y0 = (t >> 3) * TILE_Y;     // 16 tile-rows
  const int x0 = (t & 7) * TILE_X;      // 8 tile-cols
  const float* base1 = img1 + (size_t)p * (IMG_W * IMG_H);
  const float* base2 = img2 + (size_t)p * (IMG_W * IMG_H);

  // ---- gaussian window, exactly matching reference formula ----
  if (tid == 0) {
    float tmp[11];
    float s = 0.f;
    #pragma unroll
    for (int i = 0; i < 11; ++i) {
      float x = (float)(i - 5);
      tmp[i] = expf(-(x * x) / (2.0f * 1.5f * 1.5f));
      s += tmp[i];
    }
    #pragma unroll
    for (int i = 0; i < 11; ++i) gtab[i] = tmp[i] / s;
  }

  // ---- stage inputs: async direct-to-LDS for interior tiles, scalar
  //      zero-padded path for border tiles (matches conv zero padding) ----
  const bool interior = (y0 >= 32) && (y0 <= 448) && (x0 >= 64) && (x0 <= 384);
  if (interior) {
    const int nchunks = SROWS * (SW / 4);     // 44 * 22 = 968 x 16B per image
    for (int idx = tid; idx < nchunks; idx += 256) {
      const int i  = idx / (SW / 4);
      const int jq = idx - i * (SW / 4);
      const size_t goff = (size_t)(y0 - 5 + i) * IMG_W + (x0 - XOFF) + jq * 4;
      const int   loff = i * SW + jq * 4;
      async_copy16(base1 + goff, &S1[loff]);
      async_copy16(base2 + goff, &S2[loff]);
    }
    wait_async();
  } else {
    for (int idx = tid; idx < SROWS * SW; idx += 256) {
      const int i  = idx / SW;
      const int j  = idx - i * SW;
      const int gy = y0 - 5 + i;
      const int gx = x0 - XOFF + j;
      const bool ok = (gy >= 0) && (gy < IMG_H) && (gx >= 0) && (gx < IMG_W);
      const int off = gy * IMG_W + gx;
      S1[idx] = ok ? base1[off] : 0.f;
      S2[idx] = ok ? base2[off] : 0.f;
    }
  }
  __syncthreads();

  // ---- per-wave 16x16 output tile (8 waves = 2x4 subtiles of 32x64) ----
  const int wave  = tid >> 5;
  const int lane  = tid & 31;
  const int yn    = lane & 15;          // A: M index / B: N index
  const int hi    = lane >> 4;          // high half-wave
  const int rb    = (wave >> 2) * 16;   // subtile row origin in staged coords
  const int cb    = (wave & 3) * 16;    // subtile col origin
  const int wbase = wave * (16 * 32);   // per-wave scratch region

  // Band coefficients. A-vertical element (M=yn, K=4kk+v+2hi) = g[K-M];
  // B-horizontal element (K=4kk+v+2hi, N=yn) = g[K-N]. Identical per lane.
  float W[7][2];
  #pragma unroll
  for (int kk = 0; kk < 7; ++kk) {
    #pragma unroll
    for (int v = 0; v < 2; ++v) {
      const int d = 4 * kk + v + 2 * hi - yn;
      W[kk][v] = (d >= 0 && d <= 10) ? gtab[d] : 0.f;
    }
  }

  // Five convolved fields: mu1, mu2, E[x^2], E[y^2], E[xy]
  v8f O[5];
  #pragma unroll
  for (int f = 0; f < 5; ++f) {
    // ---- vertical pass: T(16 x 26pad32) = band(16x28) x S(28 x 26) ----
    #pragma unroll
    for (int half = 0; half < 2; ++half) {
      v8f acc = {0.f, 0.f, 0.f, 0.f, 0.f, 0.f, 0.f, 0.f};
      #pragma unroll
      for (int kk = 0; kk < 7; ++kk) {
        const int r0 = 4 * kk + 2 * hi;                 // B row K for vgpr 0
        const int c  = cb + 16 * half + yn + (XOFF - 5);// B col N (+3 shift)
        const float i1a = S1[(rb + r0) * SW + c];
        const float i1b = S1[(rb + r0 + 1) * SW + c];
        const float i2a = S2[(rb + r0) * SW + c];
        const float i2b = S2[(rb + r0 + 1) * SW + c];
        float b0, b1;
        if      (f == 0) { b0 = i1a;        b1 = i1b; }
        else if (f == 1) { b0 = i2a;        b1 = i2b; }
        else if (f == 2) { b0 = i1a * i1a;  b1 = i1b * i1b; }
        else if (f == 3) { b0 = i2a * i2a;  b1 = i2b * i2b; }
        else             { b0 = i1a * i2a;  b1 = i1b * i2b; }
        v2f a = {W[kk][0], W[kk][1]};
        v2f b = {b0, b1};
        acc = wmma4(a, b, acc);
      }
      // spill T (C-layout) to per-wave scratch, row-major 16x32
      #pragma unroll
      for (int e = 0; e < 8; ++e)
        scrT[wbase + (e + 8 * hi) * 32 + 16 * half + yn] = acc[e];
    }

    // ---- horizontal pass: O(16x16) = T(16x28) x band(28x16) ----
    v8f o = {0.f, 0.f, 0.f, 0.f, 0.f, 0.f, 0.f, 0.f};
    #pragma unroll
    for (int kk = 0; kk < 7; ++kk) {
      const float a0 = scrT[wbase + yn * 32 + 4 * kk + 2 * hi];
      const float a1 = scrT[wbase + yn * 32 + 4 * kk + 1 + 2 * hi];
      v2f a = {a0, a1};
      v2f b = {W[kk][0], W[kk][1]};
      o = wmma4(a, b, o);
    }
    O[f] = o;
  }

  // ---- SSIM elementwise math + per-lane sum (all pixels in-bounds) ----
  float lsum = 0.f;
  #pragma unroll
  for (int e = 0; e < 8; ++e) {
    const float mu1 = O[0][e];
    const float mu2 = O[1][e];
    const float s11 = O[2][e] - mu1 * mu1;
    const float s22 = O[3][e] - mu2 * mu2;
    const float s12 = O[4][e] - mu1 * mu2;
    const float cs  = (2.f * s12 + SSIM_C2) / (s11 + s22 + SSIM_C2);
    const float ss  = (2.f * mu1 * mu2 + SSIM_C1) /
                      (mu1 * mu1 + mu2 * mu2 + SSIM_C1) * cs;
    lsum += ss;
  }

  // ---- deterministic block reduction ----
  red[tid] = lsum;
  __syncthreads();
  for (int off = 128; off > 0; off >>= 1) {
    if (tid < off) red[tid] += red[tid + off];
    __syncthreads();
  }
  if (tid == 0) partial[blockIdx.x] = red[0];
}

__global__ __launch_bounds__(256) void ssim_reduce_kernel(
    const float* __restrict__ partial, float* __restrict__ out,
    int n, float scale) {
  __shared__ float red[256];
  float s = 0.f;
  for (int i = threadIdx.x; i < n; i += 256) s += partial[i];
  red[threadIdx.x] = s;
  __syncthreads();
  for (int off = 128; off > 0; off >>= 1) {
    if (threadIdx.x < off) red[threadIdx.x] += red[threadIdx.x + off];
    __syncthreads();
  }
  if (threadIdx.x == 0) out[0] = red[0] * scale;
}

extern "C" void kernel_launch(void* const* d_in, const int* in_sizes, int n_in,
                              void* d_out, int out_size, void* d_ws, size_t ws_size,
                              hipStream_t stream) {
  const float* img1 = (const float*)d_in[0];
  const float* img2 = (const float*)d_in[1];
  float* partial = (float*)d_ws;        // 6144 fp32 partials
  const int nblocks = 48 * 128;         // 16*3 planes x 128 tiles
  ssim_conv_kernel<<<nblocks, 256, 0, stream>>>(img1, img2, partial);
  ssim_reduce_kernel<<<1, 256, 0, stream>>>(partial, (float*)d_out, nblocks,
                                            1.0f / 12582912.0f);
}